// skipConnection_49400713838933
// MI455X (gfx1250) — compile-verified
//
#include <hip/hip_runtime.h>

// out[i] = in[2i]*w + in[2i+1]*(1-w)   -- pure HBM streaming, fp32.
// Roofline: 12 B + ~3 FLOP per output -> HBM-bound; WMMA intentionally unused.
// 16 outputs per thread: 8x global_load_b128 (NT) -> pk_mul + pairwise add
// -> 4x global_store_b128 (NT).

typedef float v4f __attribute__((ext_vector_type(4)));

#define OUTS_PER_THREAD 16
#define BLOCK_THREADS 256

__global__ __launch_bounds__(BLOCK_THREADS)
void skipConnection_stream_kernel(const float* __restrict__ in,
                                  const float* __restrict__ wp,
                                  float* __restrict__ out,
                                  long long n) {
    // Uniform address -> scalar load; w, w1 live in SGPRs.
    const float w  = wp[0];
    const float w1 = 1.0f - w;
    const v4f  wv = {w, w1, w, w1};

    const long long tid  = (long long)blockIdx.x * BLOCK_THREADS + threadIdx.x;
    const long long base = tid * OUTS_PER_THREAD;

    if (base + OUTS_PER_THREAD <= n) {
        const v4f* __restrict__ in4  = (const v4f*)(in + 2 * base);
        v4f*       __restrict__ out4 = (v4f*)(out + base);

        // 8 B128 loads in one clause: 128 B of input per thread in flight.
        v4f p[8];
#pragma unroll
        for (int j = 0; j < 8; ++j)
            p[j] = __builtin_nontemporal_load(in4 + j);

        // Each p[j] = {a0, b0, a1, b1} -> t = {a0*w, b0*w1, a1*w, b1*w1}
        // (contiguous vector mul -> v_pk_mul_f32, no lane shuffles), then
        // pairwise adds write straight into the store registers.
        v4f r[4];
#pragma unroll
        for (int j = 0; j < 4; ++j) {
            v4f t0 = p[2 * j + 0] * wv;
            v4f t1 = p[2 * j + 1] * wv;
            r[j].x = t0.x + t0.y;
            r[j].y = t0.z + t0.w;
            r[j].z = t1.x + t1.y;
            r[j].w = t1.z + t1.w;
        }

#pragma unroll
        for (int j = 0; j < 4; ++j)
            __builtin_nontemporal_store(r[j], out4 + j);
    } else {
        // Scalar tail (dead for N = 16M = 16 * 256 * 4096, kept for generality).
        for (long long i = base; i < n; ++i) {
            const float a = in[2 * i];
            const float b = in[2 * i + 1];
            out[i] = a * w + b * w1;
        }
    }
}

extern "C" void kernel_launch(void* const* d_in, const int* in_sizes, int n_in,
                              void* d_out, int out_size, void* d_ws, size_t ws_size,
                              hipStream_t stream) {
    (void)n_in; (void)d_ws; (void)ws_size; (void)in_sizes;

    const float* in = (const float*)d_in[0];   // (N, 2) fp32, interleaved
    const float* wp = (const float*)d_in[1];   // (1, 1) fp32 scalar weight
    float* out      = (float*)d_out;           // (N, 1) fp32

    const long long n = (long long)out_size;   // number of output rows
    const long long outs_per_block = (long long)BLOCK_THREADS * OUTS_PER_THREAD;
    const long long grid = (n + outs_per_block - 1) / outs_per_block;

    skipConnection_stream_kernel<<<(unsigned)grid, BLOCK_THREADS, 0, stream>>>(
        in, wp, out, n);
}